// WongWangMultiClassDecision_40939628265823
// MI455X (gfx1250) — compile-verified
//
#include <hip/hip_runtime.h>
#include <stdint.h>

// Wong-Wang multi-class decision forward pass on gfx1250.
// Mapping: 1 lane per (batch, class): tid = b*4 + c, 65536 lanes total.
// eps is streamed HBM -> LDS with the Tensor Data Mover (TENSORcnt),
// triple-buffered; the serial recurrence runs out of registers.

#define N_CLASSES 4
#define BATCH     16384
#define T_STEPS   500
#define BTOT      (BATCH * N_CLASSES)   // 65536 (row length in floats per time step)
#define BLOCK     256
#define CHUNK     16                    // time steps per LDS buffer
#define NCHUNKS   ((T_STEPS + CHUNK - 1) / CHUNK)  // 32 (last chunk = 4 steps)
#define NBUF      3                     // triple buffer -> 2 chunks in flight
#define BUF_BYTES (CHUNK * BLOCK * 4)   // 16 KB per buffer

typedef uint32_t u32x4 __attribute__((ext_vector_type(4)));
typedef uint32_t u32x8 __attribute__((ext_vector_type(8)));

// Issue one TDM 2D tile load: tile = [nsteps rows] x [256 floats],
// row stride in the tensor = BTOT floats. Descriptor per ISA cdna5 8.3/8.4.
__device__ __forceinline__ void tdm_stage(uint64_t gaddr, uint32_t lds_addr,
                                          uint32_t nsteps) {
  u32x4 g0;
  g0[0] = 1u;                                  // count=1, user mode, no gather
  g0[1] = lds_addr;                            // LDS byte address of tile start
  g0[2] = (uint32_t)gaddr;                     // global_addr[31:0]
  g0[3] = ((uint32_t)(gaddr >> 32) & 0x01FFFFFFu) | 0x80000000u; // addr[56:32] | type=2

  u32x8 g1;
  g1[0] = 0x00020000u;                         // wg_mask=0, data_size=2 (4B), no abar/iter/pad
  g1[1] = 0u;                                  // abar addr=0; tensor_dim0[15:0]=0 (65536)
  g1[2] = ((uint32_t)T_STEPS << 16) | 1u;      // tensor_dim1[15:0]=500 | tensor_dim0[31:16]=1
  g1[3] = ((uint32_t)BLOCK << 16);             // tile_dim0=256 | tensor_dim1[31:16]=0
  g1[4] = nsteps;                              // tile_dim1=nsteps, tile_dim2=0
  g1[5] = (uint32_t)BTOT;                      // tensor_dim0_stride[31:0]=65536
  g1[6] = 0u;                                  // stride hi / dim1_stride lo (unused for 2D)
  g1[7] = 0u;

  // 2D tensor: VADDR2/VADDR3 omitted (NULL).
  asm volatile("tensor_load_to_lds %0, %1" :: "s"(g0), "s"(g1) : "memory");
}

__global__ __launch_bounds__(BLOCK) void wong_wang_kernel(
    const float* __restrict__ input_signal,  // [B,4]
    const float* __restrict__ eps0,          // [B,4]
    const float* __restrict__ eps,           // [T,B,4]
    const float* __restrict__ Jm,            // [4,4]
    const float* __restrict__ pJext,
    const float* __restrict__ pI0,
    const float* __restrict__ pNa,
    const float* __restrict__ pThr,
    float* __restrict__ out)                 // [B,4]
{
  __shared__ float ldsbuf[NBUF * CHUNK * BLOCK];  // 48 KB

  const int i   = threadIdx.x;
  const int tid = blockIdx.x * BLOCK + i;
  const int c   = tid & 3;

  const float Jext = pJext[0];
  const float I0   = pI0[0];
  const float na   = pNa[0];
  const float thr  = pThr[0];

  // decay = exp(-DT/tau_ampa) = exp(-5); sigma = na*sqrt((1-exp(-10))/2)
  const float decay = __expf(-5.0f);
  const float sigma = na * sqrtf((1.0f - __expf(-10.0f)) * 0.5f);

  // x[c] = sum_k s[k]*J[k][c]; neighbor reached via shfl_xor(mask) has class c^mask
  const float j0 = Jm[c * 4 + c];
  const float j1 = Jm[(c ^ 1) * 4 + c];
  const float j2 = Jm[(c ^ 2) * 4 + c];
  const float j3 = Jm[(c ^ 3) * 4 + c];

  float s      = 0.1f;
  float inoise = eps0[tid] * na;
  const float base = I0 + Jext * input_signal[tid];  // stimulus on every step

  int  dec   = T_STEPS - 1;
  bool found = false;

  const uint64_t eps_addr = (uint64_t)(uintptr_t)eps;
  const uint32_t lds_base = (uint32_t)(uintptr_t)(&ldsbuf[0]);  // low 32 bits = LDS offset
  const uint64_t blk_byte = (uint64_t)(blockIdx.x * BLOCK) * 4u;

  // Prime the pipeline: chunks 0 and 1 in flight (wave 0 drives the TDM).
  if (i < 32) {
    tdm_stage(eps_addr + blk_byte, lds_base, CHUNK);
    tdm_stage(eps_addr + (uint64_t)CHUNK * (BTOT * 4u) + blk_byte,
              lds_base + BUF_BYTES, CHUNK);
  }

  for (int k = 0; k < NCHUNKS; ++k) {
    const int t0     = k * CHUNK;
    const int nsteps = min(CHUNK, T_STEPS - t0);

    if (i < 32) {
      if (k + 2 < NCHUNKS) {
        // Issue chunk k+2 into buffer (k+2)%3 (its readers passed last barrier),
        // then wait until only the 2 newer chunks remain -> chunk k is resident.
        const int t2 = t0 + 2 * CHUNK;
        tdm_stage(eps_addr + (uint64_t)t2 * (BTOT * 4u) + blk_byte,
                  lds_base + (uint32_t)(((k + 2) % NBUF) * BUF_BYTES),
                  (uint32_t)min(CHUNK, T_STEPS - t2));
        __builtin_amdgcn_s_wait_tensorcnt(2);
      } else if (k + 1 < NCHUNKS) {
        __builtin_amdgcn_s_wait_tensorcnt(1);
      } else {
        __builtin_amdgcn_s_wait_tensorcnt(0);
      }
    }
    __syncthreads();  // chunk k visible to all waves

    const float* __restrict__ buf = &ldsbuf[(k % NBUF) * CHUNK * BLOCK];

#pragma unroll 4
    for (int l = 0; l < nsteps; ++l) {
      const float e  = buf[l * BLOCK + i];
      const float s1 = __shfl_xor(s, 1, 32);
      const float s2 = __shfl_xor(s, 2, 32);
      const float s3 = __shfl_xor(s, 3, 32);

      const float x   = base + inoise + s * j0 + s1 * j1 + s2 * j2 + s3 * j3;
      const float ax  = fmaf(270.0f, x, -108.0f);
      const float den = 1.0f - __expf(-0.154f * ax) + 1e-6f;
      const float H   = fmaxf(ax / den, 0.0f);
      // dsdt = -s/100 + (1-s)*H*0.641/1000
      const float dsdt = fmaf(-0.01f, s, (1.0f - s) * H * 0.000641f);

      inoise = fmaf(inoise, decay, sigma * e);
      s      = fmaf(dsdt, 10.0f, s);  // s + dsdt*DT

      if (!found && s > thr) { dec = t0 + l; found = true; }
    }

    __syncthreads();  // all readers done before this buffer is re-staged
  }

  // decision_time = dec * DT / 1000 = dec * 0.01
  out[tid] = 0.01f * (float)dec;
}

extern "C" void kernel_launch(void* const* d_in, const int* in_sizes, int n_in,
                              void* d_out, int out_size, void* d_ws, size_t ws_size,
                              hipStream_t stream) {
  (void)in_sizes; (void)n_in; (void)out_size; (void)d_ws; (void)ws_size;
  const float* input_signal = (const float*)d_in[0];
  const float* eps0         = (const float*)d_in[1];
  const float* eps          = (const float*)d_in[2];
  const float* Jm           = (const float*)d_in[3];
  const float* pJext        = (const float*)d_in[4];
  const float* pI0          = (const float*)d_in[5];
  const float* pNa          = (const float*)d_in[6];
  const float* pThr         = (const float*)d_in[7];
  float* out = (float*)d_out;

  dim3 grid(BTOT / BLOCK);  // 256 blocks
  dim3 block(BLOCK);        // 8 waves each (wave32)
  wong_wang_kernel<<<grid, block, 0, stream>>>(input_signal, eps0, eps, Jm,
                                               pJext, pI0, pNa, pThr, out);
}